// GCN_66005057405276
// MI455X (gfx1250) — compile-verified
//
#include <hip/hip_runtime.h>

typedef __attribute__((ext_vector_type(2))) float v2f;
typedef __attribute__((ext_vector_type(8))) float v8f;

#define GCN_F_IN 128
#define GCN_HID  128
#define GCN_NCLS 64
#define GCN_EPS  1e-6f

// ---------------- degree / norm ----------------

__global__ __launch_bounds__(256) void gcn_deg_init(float* __restrict__ deg, int n) {
    int i = blockIdx.x * blockDim.x + threadIdx.x;
    if (i < n) deg[i] = 1.0f;  // self-loop contributes 1 to every node's degree
}

__global__ __launch_bounds__(256) void gcn_deg_count(const int* __restrict__ dst,
                                                     float* __restrict__ deg, int e) {
    int i = blockIdx.x * blockDim.x + threadIdx.x;
    if (i < e) atomicAdd(&deg[dst[i]], 1.0f);
}

__global__ __launch_bounds__(256) void gcn_dinv(float* __restrict__ deg, int n) {
    int i = blockIdx.x * blockDim.x + threadIdx.x;
    if (i < n) {
        float d = deg[i];
        deg[i] = d > 0.0f ? rsqrtf(d) : 0.0f;  // in-place: deg -> dinv
    }
}

// ---------------- fp32 WMMA GEMM: C[M,OUT] = A[M,128] * B[128,OUT] ----------------
// Block = 128 threads (4 wave32), one 16-row M-tile per block.
// Wave w computes TPW 16-col N-tiles. K=128 unrolled in 16x16x4 f32 WMMA steps.

template <int OUT, int TPW>
__global__ __launch_bounds__(128) void gcn_gemm_wmma(const float* __restrict__ A,
                                                     const float* __restrict__ B,
                                                     float* __restrict__ C, int M) {
    const int lane  = threadIdx.x & 31;
    const int wave  = threadIdx.x >> 5;
    const int mbase = blockIdx.x * 16;
    const int l16   = lane & 15;
    const int khalf = (lane >> 4) * 2;  // lanes 0-15 -> K 0/1, lanes 16-31 -> K 2/3

    int arow = mbase + l16;
    if (arow > M - 1) arow = M - 1;  // clamp (N divisible by 16, clamp is safety only)
    const float* __restrict__ Ap = A + (long)arow * GCN_F_IN;

#pragma unroll
    for (int t = 0; t < TPW; ++t) {
        const int n0 = (wave * TPW + t) * 16;
        v8f c = {};
#pragma unroll
        for (int k0 = 0; k0 < 128; k0 += 4) {
            v2f a, b;
            a.x = Ap[k0 + khalf];
            a.y = Ap[k0 + khalf + 1];
            b.x = B[(k0 + khalf) * OUT + n0 + l16];
            b.y = B[(k0 + khalf + 1) * OUT + n0 + l16];
            c = __builtin_amdgcn_wmma_f32_16x16x4_f32(
                    false, a, false, b, (short)0, c, false, false);
        }
        const int mrow = mbase + (lane >> 4) * 8;
        float* __restrict__ Cp = C + n0 + l16;
#pragma unroll
        for (int v = 0; v < 8; ++v) {
            int r = mrow + v;
            if (r < M) Cp[(long)r * OUT] = c[v];
        }
    }
}

// ---------------- aggregation ----------------
// init: out[i,f] = h[i,f]*dinv[i]^2 + bias[f]   (self-loop term + bias)
template <int F>
__global__ __launch_bounds__(256) void gcn_agg_init(const float* __restrict__ h,
                                                    const float* __restrict__ dinv,
                                                    const float* __restrict__ bias,
                                                    float* __restrict__ out, int n) {
    long i = (long)blockIdx.x * blockDim.x + threadIdx.x;
    if (i >= (long)n * F) return;
    int node = (int)(i / F);
    int feat = (int)(i % F);
    float di = dinv[node];
    out[i] = h[i] * di * di + bias[feat];
}

// edges: out[dst] += h[src] * dinv[src]*dinv[dst]; F/4 lanes per edge, float4 gather
template <int F>
__global__ __launch_bounds__(256) void gcn_agg_edges(const float* __restrict__ h,
                                                     const int* __restrict__ src,
                                                     const int* __restrict__ dst,
                                                     const float* __restrict__ dinv,
                                                     float* __restrict__ out, int e) {
    const int LPE = F / 4;
    long tid  = (long)blockIdx.x * blockDim.x + threadIdx.x;
    int edge  = (int)(tid / LPE);
    int part  = (int)(tid % LPE);
    if (edge >= e) return;
    int s = src[edge];
    int d = dst[edge];
    float norm = dinv[s] * dinv[d];
    const float4 hv = *(const float4*)(h + (long)s * F + part * 4);
    float* __restrict__ o = out + (long)d * F + part * 4;
    atomicAdd(o + 0, hv.x * norm);
    atomicAdd(o + 1, hv.y * norm);
    atomicAdd(o + 2, hv.z * norm);
    atomicAdd(o + 3, hv.w * norm);
}

// ---------------- LayerNorm(128) + ReLU, one wave32 per node ----------------

__global__ __launch_bounds__(256) void gcn_ln_relu(const float* __restrict__ in,
                                                   const float* __restrict__ gamma,
                                                   const float* __restrict__ beta,
                                                   float* __restrict__ outp, int n) {
    int lane = threadIdx.x & 31;
    int node = (int)(((long)blockIdx.x * blockDim.x + threadIdx.x) >> 5);
    if (node >= n) return;
    const float4 v = *(const float4*)(in + (long)node * 128 + lane * 4);

    float s = v.x + v.y + v.z + v.w;
#pragma unroll
    for (int off = 16; off > 0; off >>= 1) s += __shfl_xor(s, off, 32);
    float mu = s * (1.0f / 128.0f);

    float dx = v.x - mu, dy = v.y - mu, dz = v.z - mu, dw = v.w - mu;
    float sq = dx * dx + dy * dy + dz * dz + dw * dw;
#pragma unroll
    for (int off = 16; off > 0; off >>= 1) sq += __shfl_xor(sq, off, 32);
    float rstd = rsqrtf(sq * (1.0f / 128.0f) + GCN_EPS);

    const float4 g = *(const float4*)(gamma + lane * 4);
    const float4 b = *(const float4*)(beta + lane * 4);
    float4 o;
    o.x = fmaxf(dx * rstd * g.x + b.x, 0.0f);
    o.y = fmaxf(dy * rstd * g.y + b.y, 0.0f);
    o.z = fmaxf(dz * rstd * g.z + b.z, 0.0f);
    o.w = fmaxf(dw * rstd * g.w + b.w, 0.0f);
    *(float4*)(outp + (long)node * 128 + lane * 4) = o;
}

// ---------------- launch ----------------

extern "C" void kernel_launch(void* const* d_in, const int* in_sizes, int n_in,
                              void* d_out, int out_size, void* d_ws, size_t ws_size,
                              hipStream_t stream) {
    const float* x   = (const float*)d_in[0];
    const int*   ei  = (const int*)d_in[1];
    const float* W1  = (const float*)d_in[2];
    const float* b1  = (const float*)d_in[3];
    const float* g1  = (const float*)d_in[4];
    const float* be1 = (const float*)d_in[5];
    const float* W2  = (const float*)d_in[6];
    const float* b2  = (const float*)d_in[7];
    float* out = (float*)d_out;

    const int n = in_sizes[0] / GCN_F_IN;  // 50000
    const int e = in_sizes[1] / 2;         // 800000
    const int* src = ei;
    const int* dst = ei + e;

    float* ws   = (float*)d_ws;
    float* dinv = ws;                            // n
    float* h1   = dinv + n;                      // n*128
    float* a1   = h1 + (size_t)n * GCN_HID;      // n*128 (agg1, then LN in-place)
    float* h2   = a1 + (size_t)n * GCN_HID;      // n*64

    // degree / normalization
    gcn_deg_init<<<(n + 255) / 256, 256, 0, stream>>>(dinv, n);
    gcn_deg_count<<<(e + 255) / 256, 256, 0, stream>>>(dst, dinv, e);
    gcn_dinv<<<(n + 255) / 256, 256, 0, stream>>>(dinv, n);

    // layer 1: h1 = x@W1 ; a1 = A_hat@h1 + b1 ; a1 = relu(LN(a1))
    gcn_gemm_wmma<GCN_HID, 2><<<(n + 15) / 16, 128, 0, stream>>>(x, W1, h1, n);
    gcn_agg_init<GCN_HID><<<(int)(((long)n * GCN_HID + 255) / 256), 256, 0, stream>>>(
        h1, dinv, b1, a1, n);
    gcn_agg_edges<GCN_HID><<<(int)(((long)e * (GCN_HID / 4) + 255) / 256), 256, 0, stream>>>(
        h1, src, dst, dinv, a1, e);
    gcn_ln_relu<<<(int)(((long)n * 32 + 255) / 256), 256, 0, stream>>>(a1, g1, be1, a1, n);

    // layer 2: h2 = a1@W2 ; out = A_hat@h2 + b2
    gcn_gemm_wmma<GCN_NCLS, 1><<<(n + 15) / 16, 128, 0, stream>>>(a1, W2, h2, n);
    gcn_agg_init<GCN_NCLS><<<(int)(((long)n * GCN_NCLS + 255) / 256), 256, 0, stream>>>(
        h2, dinv, b2, out, n);
    gcn_agg_edges<GCN_NCLS><<<(int)(((long)e * (GCN_NCLS / 4) + 255) / 256), 256, 0, stream>>>(
        h2, src, dst, dinv, out, e);
}